// NTMMemory_62775241999226
// MI455X (gfx1250) — compile-verified
//
#include <hip/hip_runtime.h>

#define NROWS 65536
#define MCOLS 512
#define EPSF 1e-16f

typedef __attribute__((ext_vector_type(2))) float v2f;
typedef __attribute__((ext_vector_type(4))) float v4f;
typedef __attribute__((ext_vector_type(8))) float v8f;

#if defined(__has_builtin)
#  if __has_builtin(__builtin_amdgcn_wmma_f32_16x16x4_f32)
#    define HAVE_WMMA_F32 1
#  else
#    define HAVE_WMMA_F32 0
#  endif
#else
#  define HAVE_WMMA_F32 0
#endif

__device__ __forceinline__ float blockReduceSum256(float v, float* s_red) {
    const int tid = threadIdx.x;
    s_red[tid] = v;
    __syncthreads();
    for (int off = 128; off > 0; off >>= 1) {
        if (tid < off) s_red[tid] += s_red[tid + off];
        __syncthreads();
    }
    float r = s_red[0];
    __syncthreads();
    return r;
}

// ---------------------------------------------------------------------------
// K1: content addressing. One wave handles 16 rows; dot(mem_row+eps, k+eps)
// via V_WMMA_F32_16X16X4_F32. B replicates the key chunk in all 16 columns so
// every column of D equals the row-dot. K-slot relabeling trick: lo lanes own
// K-slots {kb,kb+1}, hi lanes own {kb+4,kb+5} (wmma1) / {kb+2,kb+3} and
// {kb+6,kb+7} (wmma2) — consistent between A and B, so no cross-lane shuffles
// are needed at all. Row norms accumulate on the VALU side.
// ---------------------------------------------------------------------------
__global__ __launch_bounds__(256) void k_content(const float* __restrict__ mem,
                                                 const float* __restrict__ kvec,
                                                 const float* __restrict__ beta_p,
                                                 float* __restrict__ expw,
                                                 float* __restrict__ partial1) {
    __shared__ __align__(16) float s_kk[MCOLS];
    __shared__ float s_dot[8][16];
    __shared__ float s_red[256];
    const int tid = threadIdx.x;

    // Stage kk = k + eps into LDS (coalesced), compute ||kk||.
    for (int j = tid; j < MCOLS; j += 256) s_kk[j] = kvec[j] + EPSF;
    __syncthreads();
    float kn = s_kk[tid] * s_kk[tid] + s_kk[tid + 256] * s_kk[tid + 256];
    const float knorm = sqrtf(blockReduceSum256(kn, s_red));
    const float beta = beta_p[0];

    const int lane = tid & 31;
    const int wave = tid >> 5;
    const int rowbase = blockIdx.x * 128 + wave * 16;
    const int m = lane & 15;
    const bool hi = lane >= 16;
    // lo lanes walk mem[row][kb..kb+3]; hi lanes walk mem[row][kb+4..kb+7]
    const float* rowp = mem + (size_t)(rowbase + m) * MCOLS + (hi ? 4 : 0);
    const float* kkp = &s_kk[hi ? 4 : 0];

    float nacc = 0.f;
#if HAVE_WMMA_F32
    v8f c = {0.f, 0.f, 0.f, 0.f, 0.f, 0.f, 0.f, 0.f};
#else
    float dacc = 0.f;
#endif
    v4f v = *(const v4f*)(rowp);  // software pipeline: preload first chunk
    for (int kb = 0; kb < MCOLS; kb += 8) {
        v4f vn = v;
        if (kb + 8 < MCOLS) vn = *(const v4f*)(rowp + kb + 8);  // next chunk in flight
        v4f vv = v + EPSF;
        nacc += vv.x * vv.x + vv.y * vv.y + vv.z * vv.z + vv.w * vv.w;
        const v4f q = *(const v4f*)(kkp + kb);  // kk[kb..+3] (lo) / kk[kb+4..+7] (hi)
#if HAVE_WMMA_F32
        v2f a1, b1, a2, b2;
        a1.x = vv.x; a1.y = vv.y;  b1.x = q.x; b1.y = q.y;
        a2.x = vv.z; a2.y = vv.w;  b2.x = q.z; b2.y = q.w;
        c = __builtin_amdgcn_wmma_f32_16x16x4_f32(false, a1, false, b1,
                                                  (short)0, c, false, false);
        c = __builtin_amdgcn_wmma_f32_16x16x4_f32(false, a2, false, b2,
                                                  (short)0, c, false, false);
#else
        dacc += vv.x * q.x + vv.y * q.y + vv.z * q.z + vv.w * q.w;
#endif
        v = vn;
    }
    nacc += __shfl_xor(nacc, 16);  // lanes 0-15 now hold full row norm^2

#if HAVE_WMMA_F32
    // D layout: vgpr m holds dot(row m) on lanes 0-15 and dot(row m+8) on 16-31.
    if (lane == 0) {
#pragma unroll
        for (int mm = 0; mm < 8; ++mm) s_dot[wave][mm] = c[mm];
    }
    if (lane == 16) {
#pragma unroll
        for (int mm = 0; mm < 8; ++mm) s_dot[wave][8 + mm] = c[mm];
    }
    __syncthreads();
#else
    dacc += __shfl_xor(dacc, 16);
    if (lane < 16) s_dot[wave][lane] = dacc;
    __syncthreads();
#endif

    float ew = 0.f;
    if (lane < 16) {
        const float dot = s_dot[wave][lane];
        const float cosv = dot / (sqrtf(nacc) * knorm + EPSF);
        ew = expf(beta * cosv);
        expw[rowbase + lane] = ew;
    }
    float esum = ew;  // lanes >= 16 contribute 0 within their own xor group
    esum += __shfl_xor(esum, 8);
    esum += __shfl_xor(esum, 4);
    esum += __shfl_xor(esum, 2);
    esum += __shfl_xor(esum, 1);
    __syncthreads();
    if (lane == 0) s_red[wave] = esum;
    __syncthreads();
    if (tid == 0) {
        float t = 0.f;
#pragma unroll
        for (int wv = 0; wv < 8; ++wv) t += s_red[wv];
        partial1[blockIdx.x] = t;
    }
}

// K2: reduce 512 block partials -> softmax denominator.
__global__ __launch_bounds__(256) void k_reduce512(const float* __restrict__ part,
                                                   float* __restrict__ scal) {
    __shared__ float s_red[256];
    float v = part[threadIdx.x] + part[threadIdx.x + 256];
    float s = blockReduceSum256(v, s_red);
    if (threadIdx.x == 0) scal[0] = s;
}

// K3: wg[i] = g * softmax_i + (1-g) * w_prev[i]
__global__ __launch_bounds__(256) void k_interp(const float* __restrict__ expw,
                                                const float* __restrict__ w_prev,
                                                const float* __restrict__ g_p,
                                                const float* __restrict__ scal,
                                                float* __restrict__ wg) {
    const int i = blockIdx.x * 256 + threadIdx.x;
    const float g = g_p[0];
    const float inv = 1.0f / scal[0];
    wg[i] = g * expw[i] * inv + (1.0f - g) * w_prev[i];
}

// K4: circular 3-tap shift + sharpen, with per-block partial sums.
__global__ __launch_bounds__(256) void k_shift_pow(const float* __restrict__ wg,
                                                   const float* __restrict__ s_p,
                                                   const float* __restrict__ gamma_p,
                                                   float* __restrict__ wp,
                                                   float* __restrict__ partial2) {
    __shared__ float s_red[256];
    const int i = blockIdx.x * 256 + threadIdx.x;
    const float s0 = s_p[0], s1 = s_p[1], s2 = s_p[2];
    const float gamma = gamma_p[0];
    const int im1 = (i + NROWS - 1) & (NROWS - 1);
    const int ip1 = (i + 1) & (NROWS - 1);
    const float sh = wg[im1] * s0 + wg[i] * s1 + wg[ip1] * s2;
    const float p = powf(sh, gamma);
    wp[i] = p;
    const float b = blockReduceSum256(p, s_red);
    if (threadIdx.x == 0) partial2[blockIdx.x] = b;
}

// K5: reduce 256 partials -> 1/(sum(wp)+eps)
__global__ __launch_bounds__(256) void k_reduce256(const float* __restrict__ part,
                                                   float* __restrict__ scal) {
    __shared__ float s_red[256];
    float s = blockReduceSum256(part[threadIdx.x], s_red);
    if (threadIdx.x == 0) scal[1] = 1.0f / (s + EPSF);
}

// K6: big fused pass. Each block owns 128 rows; thread t owns columns 2t,2t+1.
// Reads mem non-temporally (dead afterwards), writes new_mem with NT stores so
// the 192MB L2 keeps serving mem lines loaded in K1. Accumulates r partials.
__global__ __launch_bounds__(256) void k_readwrite(const float* __restrict__ mem,
                                                   const float* __restrict__ wp,
                                                   const float* __restrict__ scal,
                                                   const float* __restrict__ e,
                                                   const float* __restrict__ a,
                                                   float* __restrict__ rpart,
                                                   float* __restrict__ out) {
    const int t = threadIdx.x;
    const float inv = scal[1];
    const v2f e2 = *(const v2f*)(e + 2 * t);
    const v2f a2 = *(const v2f*)(a + 2 * t);
    const int row0 = blockIdx.x * 128;
    float r0 = 0.f, r1 = 0.f;
    float* newmem = out + MCOLS;  // out[0..511] = r, rest = new_mem
#pragma unroll 4
    for (int rr = 0; rr < 128; ++rr) {
        const int row = row0 + rr;
        const float wi = wp[row] * inv;  // broadcast load
        const size_t off = (size_t)row * MCOLS + 2 * t;
        const v2f mv = __builtin_nontemporal_load((const v2f*)(mem + off));
        r0 += wi * mv.x;
        r1 += wi * mv.y;
        v2f nv;
        nv.x = mv.x * (1.0f - wi * e2.x) + wi * a2.x;
        nv.y = mv.y * (1.0f - wi * e2.y) + wi * a2.y;
        __builtin_nontemporal_store(nv, (v2f*)(newmem + off));
    }
    rpart[(size_t)blockIdx.x * MCOLS + 2 * t] = r0;
    rpart[(size_t)blockIdx.x * MCOLS + 2 * t + 1] = r1;
}

// K7: deterministic reduction of 512 r-partials -> out[0..511]
__global__ __launch_bounds__(256) void k_rfinal(const float* __restrict__ rpart,
                                                float* __restrict__ out) {
    const int j = blockIdx.x * 256 + threadIdx.x;  // 0..511
    float s = 0.f;
    for (int p = 0; p < 512; ++p) s += rpart[(size_t)p * MCOLS + j];
    out[j] = s;
}

extern "C" void kernel_launch(void* const* d_in, const int* in_sizes, int n_in,
                              void* d_out, int out_size, void* d_ws, size_t ws_size,
                              hipStream_t stream) {
    (void)in_sizes; (void)n_in; (void)out_size; (void)ws_size;
    const float* mem    = (const float*)d_in[0];
    const float* kvec   = (const float*)d_in[1];
    const float* beta   = (const float*)d_in[2];
    const float* g      = (const float*)d_in[3];
    const float* s      = (const float*)d_in[4];
    const float* gamma  = (const float*)d_in[5];
    const float* w_prev = (const float*)d_in[6];
    const float* e      = (const float*)d_in[7];
    const float* a      = (const float*)d_in[8];
    float* out = (float*)d_out;
    float* ws  = (float*)d_ws;

    // workspace layout (floats)
    float* expw  = ws;                 // 65536
    float* wg    = ws + 65536;         // 65536
    float* wp    = ws + 131072;        // 65536
    float* p1    = ws + 196608;        // 512
    float* p2    = ws + 197120;        // 256
    float* scal  = ws + 197376;        // 2 (+pad)
    float* rpart = ws + 197440;        // 512*512

    k_content  <<<512, 256, 0, stream>>>(mem, kvec, beta, expw, p1);
    k_reduce512<<<  1, 256, 0, stream>>>(p1, scal);
    k_interp   <<<256, 256, 0, stream>>>(expw, w_prev, g, scal, wg);
    k_shift_pow<<<256, 256, 0, stream>>>(wg, s, gamma, wp, p2);
    k_reduce256<<<  1, 256, 0, stream>>>(p2, scal);
    k_readwrite<<<512, 256, 0, stream>>>(mem, wp, scal, e, a, rpart, out);
    k_rfinal   <<<  2, 256, 0, stream>>>(rpart, out);
}